// GraphConvolutionLayer_59785944760390
// MI455X (gfx1250) — compile-verified
//
#include <hip/hip_runtime.h>
#include <math.h>

#define NN 100000
#define NE 3200000
#define D 128
#define BN_EPS 1e-5f

typedef float v2f __attribute__((ext_vector_type(2)));
typedef float v8f __attribute__((ext_vector_type(8)));

// ---------------- zero workspace ----------------
__global__ void zero_kernel(float* __restrict__ p, long long n) {
    long long i = (long long)blockIdx.x * blockDim.x + threadIdx.x;
    long long stride = (long long)gridDim.x * blockDim.x;
    for (; i < n; i += stride) p[i] = 0.0f;
}

// ---------------- degree over dst ----------------
__global__ void degree_kernel(const int* __restrict__ ei, float* __restrict__ deg) {
    int e = blockIdx.x * blockDim.x + threadIdx.x;
    if (e < NE) {
        int dst = ei[NE + e];
        atomicAdd(&deg[dst], 1.0f);
    }
}

// ---------------- deg -> d^{-1/2} in place ----------------
__global__ void dinv_kernel(float* __restrict__ deg) {
    int i = blockIdx.x * blockDim.x + threadIdx.x;
    if (i < NN) {
        float d = deg[i];
        deg[i] = (d > 0.0f) ? rsqrtf(fmaxf(d, 1.0f)) : 0.0f;
    }
}

// ---------------- edge gather + normalized scatter-add of x ----------------
// one wave (32 lanes) per edge; 4 features per lane
__global__ void scatter_kernel(const float4* __restrict__ x4,
                               const int* __restrict__ ei,
                               const float* __restrict__ dinv,
                               float* __restrict__ aggx) {
    long long t = (long long)blockIdx.x * blockDim.x + threadIdx.x;
    long long e = t >> 5;
    int c = (int)(t & 31);
    if (e < NE) {
        int src = ei[e];
        int dst = ei[NE + e];
        float norm = dinv[src] * dinv[dst];
        float4 v = x4[(long long)src * 32 + c];
        float* o = aggx + (long long)dst * D + c * 4;
        atomicAdd(o + 0, v.x * norm);
        atomicAdd(o + 1, v.y * norm);
        atomicAdd(o + 2, v.z * norm);
        atomicAdd(o + 3, v.w * norm);
    }
}

// ---------------- agg = aggx @ W^T via fp32 WMMA ----------------
// block = 256 threads = 8 waves; each wave -> one 16x16 output tile.
// blockIdx.x -> 16-row stripe (6250 stripes); wave id -> 16-col tile (8 tiles).
__global__ void gemm_wmma_kernel(const float* __restrict__ aggx,
                                 const float* __restrict__ W,
                                 float* __restrict__ out) {
    int lane = threadIdx.x & 31;
    int wid  = threadIdx.x >> 5;       // 0..7 column tile
    int R = blockIdx.x * 16;           // row base
    int C = wid * 16;                  // col base
    int half = lane >> 4;              // 0: K=0,1  1: K=2,3
    int l = lane & 15;

    // A: 16x4 fp32 tile of aggx rows; B: 4x16 tile of W^T == rows of W
    const float* arow = aggx + (long long)(R + l) * D + 2 * half;
    const float* brow = W    + (long long)(C + l) * D + 2 * half;

    v8f acc = {};
    #pragma unroll 8
    for (int kk = 0; kk < D; kk += 4) {
        v2f a = *(const v2f*)(arow + kk);
        v2f b = *(const v2f*)(brow + kk);
        acc = __builtin_amdgcn_wmma_f32_16x16x4_f32(
            /*neg_a=*/false, a, /*neg_b=*/false, b,
            /*c_mod=*/(short)0, acc, /*reuse_a=*/false, /*reuse_b=*/false);
    }

    // C/D layout: VGPR v, lanes 0-15 -> M=v, lanes 16-31 -> M=v+8; N = lane%16
    float* o = out + (long long)R * D + C;
    #pragma unroll
    for (int v = 0; v < 8; v++) {
        o[(long long)(v + 8 * half) * D + l] = acc[v];
    }
}

// ---------------- BN partial sums per column ----------------
__global__ void bn_partial_kernel(const float* __restrict__ agg,
                                  float* __restrict__ bnsum,
                                  float* __restrict__ bnsqs) {
    int t = blockIdx.x * blockDim.x + threadIdx.x;
    int col = t & (D - 1);
    int r0 = t >> 7;
    int rstride = (gridDim.x * blockDim.x) >> 7;
    float s = 0.0f, q = 0.0f;
    for (int r = r0; r < NN; r += rstride) {
        float v = agg[(long long)r * D + col];
        s += v;
        q += v * v;
    }
    atomicAdd(&bnsum[col], s);
    atomicAdd(&bnsqs[col], q);
}

__global__ void bn_finalize_kernel(const float* __restrict__ bnsum,
                                   const float* __restrict__ bnsqs,
                                   float* __restrict__ stats) {
    int c = threadIdx.x;
    if (c < D) {
        float mean = bnsum[c] * (1.0f / (float)NN);
        float var = bnsqs[c] * (1.0f / (float)NN) - mean * mean;
        var = fmaxf(var, 0.0f);
        stats[c] = mean;
        stats[D + c] = rsqrtf(var + BN_EPS);
    }
}

// ---------------- normalize + ReLU + residual (in place on d_out) ----------------
__global__ void output_kernel(float* __restrict__ out,
                              const float* __restrict__ x,
                              const float* __restrict__ stats,
                              const float* __restrict__ gamma,
                              const float* __restrict__ beta) {
    long long i = (long long)blockIdx.x * blockDim.x + threadIdx.x;
    if (i < (long long)NN * D) {
        int c = (int)(i & (D - 1));
        float a = out[i];
        float bn = (a - stats[c]) * stats[D + c] * gamma[c] + beta[c];
        out[i] = fmaxf(bn, 0.0f) + x[i];
    }
}

extern "C" void kernel_launch(void* const* d_in, const int* in_sizes, int n_in,
                              void* d_out, int out_size, void* d_ws, size_t ws_size,
                              hipStream_t stream) {
    const float* x     = (const float*)d_in[0];
    const int*   ei    = (const int*)d_in[1];
    const float* W     = (const float*)d_in[2];
    const float* gamma = (const float*)d_in[3];
    const float* beta  = (const float*)d_in[4];
    float* out = (float*)d_out;

    // workspace layout (floats): deg/dinv [NN] | aggx [NN*D] | bnsum [D] | bnsqs [D] | stats [2D]
    float* deg   = (float*)d_ws;
    float* aggx  = deg + NN;
    float* bnsum = aggx + (long long)NN * D;
    float* bnsqs = bnsum + D;
    float* stats = bnsqs + D;

    // zero deg..bnsqs (contiguous)
    long long nzero = (long long)NN + (long long)NN * D + 2 * D;
    zero_kernel<<<2048, 256, 0, stream>>>((float*)d_ws, nzero);

    degree_kernel<<<(NE + 255) / 256, 256, 0, stream>>>(ei, deg);
    dinv_kernel<<<(NN + 255) / 256, 256, 0, stream>>>(deg);

    // 32 lanes per edge
    long long scat_threads = (long long)NE * 32;
    int scat_blocks = (int)((scat_threads + 255) / 256);
    scatter_kernel<<<scat_blocks, 256, 0, stream>>>((const float4*)x, ei, deg, aggx);

    // GEMM: agg = aggx @ W^T  -> d_out
    gemm_wmma_kernel<<<NN / 16, 256, 0, stream>>>(aggx, W, out);

    bn_partial_kernel<<<512, 256, 0, stream>>>(out, bnsum, bnsqs);
    bn_finalize_kernel<<<1, 128, 0, stream>>>(bnsum, bnsqs, stats);

    long long nout = (long long)NN * D;
    output_kernel<<<(int)((nout + 255) / 256), 256, 0, stream>>>(out, x, stats, gamma, beta);
}